// Adalibi_69707319214284
// MI455X (gfx1250) — compile-verified
//
#include <hip/hip_runtime.h>
#include <hip/hip_bf16.h>
#include <cstdint>

// ---------------------------------------------------------------------------
// Problem constants (from the reference file; shapes are compile-time fixed).
// ---------------------------------------------------------------------------
#define SEQ_N   2048
#define HEADS   16
#define PBINS   2
#define PN      (PBINS * SEQ_N)   // 4096 columns
// Output: (N, 1, H, P*N) float32, flat index = (n*HEADS + h)*PN + col

typedef int v4i __attribute__((ext_vector_type(4)));
using gv4i = __attribute__((address_space(1))) v4i;   // global int4
using lv4i = __attribute__((address_space(3))) v4i;   // LDS int4

// --- CDNA5 async LDS->global store path (ASYNCcnt) -------------------------
#if defined(__has_builtin)
#  if __has_builtin(__builtin_amdgcn_global_store_async_from_lds_b128)
#    define HAVE_ASYNC_BUILTIN 1
#  endif
#  if __has_builtin(__builtin_amdgcn_s_wait_asynccnt)
#    define HAVE_WAIT_BUILTIN 1
#  endif
#endif

__device__ __forceinline__ void async_store_b128(void* gdst, void* lsrc) {
#ifdef HAVE_ASYNC_BUILTIN
    __builtin_amdgcn_global_store_async_from_lds_b128(
        (gv4i*)(uintptr_t)gdst, (lv4i*)(uintptr_t)lsrc, 0, 0);
#else
    uint64_t ga = (uint64_t)(uintptr_t)gdst;       // flat addr of global mem
    uint32_t la = (uint32_t)(uintptr_t)lsrc;       // low 32 bits = LDS offset
    asm volatile("global_store_async_from_lds_b128 %0, %1, off"
                 :: "v"(ga), "v"(la) : "memory");
#endif
}

__device__ __forceinline__ void wait_asynccnt0() {
#ifdef HAVE_WAIT_BUILTIN
    __builtin_amdgcn_s_wait_asynccnt(0);
#else
    asm volatile("s_wait_asynccnt 0x0" ::: "memory");
#endif
}

// ---------------------------------------------------------------------------
// Kernel A: compute inv_p[n] = #{m in 1..n : idx_p[m] != idx_p[m-1]}
// idx_p(n) = ceil(((k+n) - u_p)/delta_p), computed in f32 like the reference.
// One block, 128 threads: p = tid/64, each of 64 chunks covers 32 elements.
// ---------------------------------------------------------------------------
__global__ __launch_bounds__(128)
void Adalibi_scan_kernel(const float* __restrict__ delta,
                         const float* __restrict__ u,
                         const int* __restrict__ kptr,
                         int* __restrict__ inv /* [2][SEQ_N] */) {
    __shared__ int chunk_sums[PBINS][64];
    const int tid = threadIdx.x;
    const int p = tid >> 6;     // 0..1
    const int c = tid & 63;     // 0..63
    const float d  = delta[p];
    const float uu = u[p];
    const int   k  = *kptr;

    const int base = c * 32;
    int s = 0;
    for (int i = 0; i < 32; ++i) {
        int n = base + i;
        if (n >= 1) {
            int i0 = (int)ceilf(((float)(k + n - 1) - uu) / d);
            int i1 = (int)ceilf(((float)(k + n)     - uu) / d);
            s += (i1 != i0) ? 1 : 0;
        }
    }
    chunk_sums[p][c] = s;
    __syncthreads();

    int run = 0;                          // exclusive prefix over chunks
    for (int j = 0; j < c; ++j) run += chunk_sums[p][j];

    for (int i = 0; i < 32; ++i) {
        int n = base + i;
        if (n >= 1) {
            int i0 = (int)ceilf(((float)(k + n - 1) - uu) / d);
            int i1 = (int)ceilf(((float)(k + n)     - uu) / d);
            run += (i1 != i0) ? 1 : 0;
        }
        inv[p * SEQ_N + n] = run;
    }
}

// ---------------------------------------------------------------------------
// Kernel B: one block per (n, h). Stage a 4096-float (16 KB) tile in LDS
// (zeros + 2 one-hot entries scaled by exp(slope_h/2)/sqrt(2)), then stream
// it to HBM with the CDNA5 async LDS->global store engine.
// Store-bound: 512 MB total -> ~22 us floor at 23.3 TB/s.
// ---------------------------------------------------------------------------
__global__ __launch_bounds__(256)
void Adalibi_fill_kernel(const int* __restrict__ inv,
                         float* __restrict__ out) {
    const int blk = blockIdx.x;          // 0 .. SEQ_N*HEADS-1
    const int n = blk >> 4;              // row
    const int h = blk & (HEADS - 1);     // head

    // H=16 is a power of two: slope_h = 2^(-0.5*(h+1)); value = exp(slope/2)/sqrt(2)
    const float slope = exp2f(-0.5f * (float)(h + 1));
    const float val   = expf(0.5f * slope) * 0.7071067811865476f;

    const int c0 = inv[n];               // non-zero col in [0, 2048)
    const int c1 = SEQ_N + inv[SEQ_N + n];

    __shared__ float tile[PN];           // 16 KB
    float4* t4 = (float4*)tile;
    const int tid = threadIdx.x;

#pragma unroll
    for (int it = 0; it < 4; ++it) {
        const int q   = tid + it * 256;  // float4 index 0..1023
        const int col = q << 2;
        float4 v = make_float4(0.f, 0.f, 0.f, 0.f);
        if (c0 >= col && c0 < col + 4) (&v.x)[c0 - col] = val;
        if (c1 >= col && c1 < col + 4) (&v.x)[c1 - col] = val;
        t4[q] = v;                       // ds_store_b128
    }
    __syncthreads();                     // LDS visible to all lanes

    float* dst = out + (size_t)blk * PN;
#pragma unroll
    for (int it = 0; it < 4; ++it) {
        const int q = tid + it * 256;
        async_store_b128(dst + (q << 2), &tile[q << 2]);  // 16B/lane, ASYNCcnt
    }
    wait_asynccnt0();
}

// ---------------------------------------------------------------------------
// Host-side launcher
// ---------------------------------------------------------------------------
extern "C" void kernel_launch(void* const* d_in, const int* in_sizes, int n_in,
                              void* d_out, int out_size, void* d_ws, size_t ws_size,
                              hipStream_t stream) {
    // Inputs (setup_inputs order): x(f32, unused), delta(f32,[P]), u(f32,[P]),
    // seq_len(int scalar, compile-time 2048), k(int scalar).
    const float* delta = (const float*)d_in[1];
    const float* u     = (const float*)d_in[2];
    const int*   kptr  = (const int*)d_in[4];
    float*       out   = (float*)d_out;
    int*         inv   = (int*)d_ws;     // 2*2048 ints of scratch

    Adalibi_scan_kernel<<<dim3(1), dim3(128), 0, stream>>>(delta, u, kptr, inv);
    Adalibi_fill_kernel<<<dim3(SEQ_N * HEADS), dim3(256), 0, stream>>>(inv, out);
}